// HierarchicalFocalLoss_3994319585385
// MI455X (gfx1250) — compile-verified
//
#include <hip/hip_runtime.h>
#include <stdint.h>

#define B_ROWS   32768
#define NCLASS   10000
#define NV4      2500      // NCLASS / 4 float4 chunks per row
#define TPB      256       // 8 wave32 per workgroup

// ---------------------------------------------------------------------------
// Kernel 1: one workgroup per row.
//   Stage the 40KB row into LDS via CDNA5 async global->LDS b128 copies
//   (ASYNCcnt), then: pass 1 = max/argmax from LDS, pass 2 = sum exp(x-m)
//   from LDS, then thread 0 computes the hierarchical focal loss for the row.
// ---------------------------------------------------------------------------
__global__ __launch_bounds__(TPB) void hfl_row_kernel(
    const float* __restrict__ logits,
    const int*   __restrict__ targets,   // (3, B)
    const int*   __restrict__ s2gf,      // (NCLASS, 2)
    float*       __restrict__ row_loss)  // (B,)
{
    __shared__ __align__(16) float srow[NCLASS];
    __shared__ float rmax[TPB];
    __shared__ int   ridx[TPB];
    __shared__ float rsum[TPB];

    const int row = blockIdx.x;
    const int tid = threadIdx.x;

    // Uniform 64-bit base of this row for the GVS-mode async loads.
    const uint64_t gbase = (uint64_t)(uintptr_t)(logits + (size_t)row * NCLASS);
    const uint32_t lds_base = (uint32_t)(uintptr_t)(&srow[0]);

    // ---- async copy: global -> LDS, 16B per lane per issue --------------
    for (int i4 = tid; i4 < NV4; i4 += TPB) {
        uint32_t lda  = lds_base + (uint32_t)i4 * 16u;  // LDS byte address
        uint32_t voff = (uint32_t)i4 * 16u;             // 32-bit global offset
        asm volatile("global_load_async_to_lds_b128 %0, %1, %2"
                     :: "v"(lda), "v"(voff), "s"(gbase) : "memory");
    }
    asm volatile("s_wait_asynccnt 0" ::: "memory");  // drain this wave's copies
    __syncthreads();                                 // make LDS visible WG-wide

    const float4* __restrict__ srow4 = (const float4*)srow;

    // ---- pass 1: per-thread max + argmax over its strided chunks --------
    float m  = -3.402823466e38f;
    int   mi = 0;
    for (int i4 = tid; i4 < NV4; i4 += TPB) {
        float4 v = srow4[i4];
        int b = i4 * 4;
        if (v.x > m) { m = v.x; mi = b;     }
        if (v.y > m) { m = v.y; mi = b + 1; }
        if (v.z > m) { m = v.z; mi = b + 2; }
        if (v.w > m) { m = v.w; mi = b + 3; }
    }
    rmax[tid] = m;
    ridx[tid] = mi;
    __syncthreads();

    // tree reduce (value, then lowest index on ties == first-occurrence argmax)
    for (int off = TPB / 2; off > 0; off >>= 1) {
        if (tid < off) {
            float v2 = rmax[tid + off]; int i2 = ridx[tid + off];
            float v1 = rmax[tid];       int i1 = ridx[tid];
            if (v2 > v1 || (v2 == v1 && i2 < i1)) { rmax[tid] = v2; ridx[tid] = i2; }
        }
        __syncthreads();
    }
    const float rm = rmax[0];   // row max, broadcast

    // ---- pass 2: sum of exp(x - m) from LDS -----------------------------
    float acc = 0.0f;
    for (int i4 = tid; i4 < NV4; i4 += TPB) {
        float4 v = srow4[i4];
        acc += __expf(v.x - rm) + __expf(v.y - rm)
             + __expf(v.z - rm) + __expf(v.w - rm);
    }
    rsum[tid] = acc;
    __syncthreads();
    for (int off = TPB / 2; off > 0; off >>= 1) {
        if (tid < off) rsum[tid] += rsum[tid + off];
        __syncthreads();
    }

    // ---- thread 0: loss for this row ------------------------------------
    if (tid == 0) {
        const float ssum = rsum[0];
        const int   pc   = ridx[0];                    // predicted class
        const int   tf = targets[0 * B_ROWS + row];    // true family
        const int   tg = targets[1 * B_ROWS + row];    // true genus
        const int   ts = targets[2 * B_ROWS + row];    // true species
        const int   pg = s2gf[pc * 2 + 0];             // predicted genus
        const int   pf = s2gf[pc * 2 + 1];             // predicted family

        const float p = __expf(srow[ts] - rm) / ssum;  // true-class softmax prob

        const bool same_sp = (pc == ts);
        const bool same_ge = (pg == tg) && !same_sp;
        const bool same_fa = (pf == tf) && !same_ge && !same_sp;
        const float pen = same_sp ? 0.0f : (same_ge ? 0.6f : (same_fa ? 0.3f : 1.0f));

        const float omp  = 1.0f - p;                   // focal weight, GAMMA=2
        const float loss = pen * omp * omp * (-__logf(p + 1e-12f));
        row_loss[row] = loss;
    }
}

// ---------------------------------------------------------------------------
// Kernel 2: deterministic mean over the 32768 per-row losses (fixed-order
// strided accumulation + fixed tree reduction; no float atomics).
// ---------------------------------------------------------------------------
__global__ __launch_bounds__(TPB) void hfl_reduce_kernel(
    const float* __restrict__ row_loss, float* __restrict__ out)
{
    __shared__ float s[TPB];
    const int tid = threadIdx.x;
    float acc = 0.0f;
    for (int i = tid; i < B_ROWS; i += TPB) acc += row_loss[i];
    s[tid] = acc;
    __syncthreads();
    for (int off = TPB / 2; off > 0; off >>= 1) {
        if (tid < off) s[tid] += s[tid + off];
        __syncthreads();
    }
    if (tid == 0) out[0] = s[0] * (1.0f / (float)B_ROWS);
}

// ---------------------------------------------------------------------------
extern "C" void kernel_launch(void* const* d_in, const int* in_sizes, int n_in,
                              void* d_out, int out_size, void* d_ws, size_t ws_size,
                              hipStream_t stream)
{
    const float* logits  = (const float*)d_in[0];   // (B, NCLASS) f32
    const int*   targets = (const int*)d_in[1];     // (3, B) i32
    const int*   s2gf    = (const int*)d_in[2];     // (NCLASS, 2) i32
    float*       out     = (float*)d_out;           // scalar f32
    float*       rloss   = (float*)d_ws;            // B floats of scratch

    hfl_row_kernel<<<B_ROWS, TPB, 0, stream>>>(logits, targets, s2gf, rloss);
    hfl_reduce_kernel<<<1, TPB, 0, stream>>>(rloss, out);
}